// spatial_transformer_85667417686182
// MI455X (gfx1250) — compile-verified
//
#include <hip/hip_runtime.h>
#include <cstdint>

// Spatial transformer (bilinear grid sample), MI455X / gfx1250.
// U:     [8, 64, 256, 256] f32
// theta: [8, 2, 65536] f32
// out:   [8, 64, 256, 256] f32
//
// Bandwidth-bound: ~273 MB total traffic -> ~11.7 us floor @ 23.3 TB/s.
// Strategy: coalesced NT stores (protect U's L2 residency), per-thread
// 64-channel loop amortizing address math, async global->LDS theta staging,
// global_prefetch of the next channel plane's rows.

#define B_     8
#define C_     64
#define HW_    256
#define N_     65536    // out_h * out_w
#define PLANE_ 65536    // H * W

__global__ __launch_bounds__(256) void st_bilinear_kernel(
    const float* __restrict__ U,
    const float* __restrict__ theta,
    float* __restrict__ out)
{
    const int tid = threadIdx.x;          // output column 0..255
    const int row = blockIdx.x;           // output row    0..255
    const int b   = blockIdx.y;           // batch         0..7
    const int n   = (row << 8) | tid;     // flat output position

    __shared__ float sth[2 * 256];

    const float* g0 = theta + (size_t)b * (2 * N_) + n;  // theta[b,0,n]
    const float* g1 = g0 + N_;                           // theta[b,1,n]

    float t0, t1;
#if defined(__gfx1250__)
    {
        // Stage theta through LDS with the gfx1250 async copy path.
        unsigned l0 = (unsigned)(uintptr_t)(&sth[tid]);
        unsigned l1 = (unsigned)(uintptr_t)(&sth[256 + tid]);
        unsigned long long a0 = (unsigned long long)(uintptr_t)g0;
        unsigned long long a1 = (unsigned long long)(uintptr_t)g1;
        asm volatile("global_load_async_to_lds_b32 %0, %1, off"
                     :: "v"(l0), "v"(a0) : "memory");
        asm volatile("global_load_async_to_lds_b32 %0, %1, off"
                     :: "v"(l1), "v"(a1) : "memory");
        asm volatile("s_wait_asynccnt 0x0" ::: "memory");
        // Each thread reads back only its own slot -> no barrier needed.
        t0 = sth[tid];
        t1 = sth[256 + tid];
    }
#else
    t0 = *g0;
    t1 = *g1;
#endif

    // Grid coords in [-1,1]; source pixel coords.
    const float gx = fmaf((float)tid, 2.0f / 255.0f, -1.0f);
    const float gy = fmaf((float)row, 2.0f / 255.0f, -1.0f);
    const float x = (t0 + gx + 1.0f) * 127.5f;   // (W-1)/2
    const float y = (t1 + gy + 1.0f) * 127.5f;   // (H-1)/2

    const float x0f = floorf(x);
    const float y0f = floorf(y);
    const int x0 = min(max((int)x0f,     0), HW_ - 2);
    const int x1 = min(max((int)x0f + 1, 0), HW_ - 1);
    const int y0 = min(max((int)y0f,     0), HW_ - 2);
    const int y1 = min(max((int)y0f + 1, 0), HW_ - 1);

    // Weights use the *clamped* integer coords, matching the reference.
    const float x0w = (float)x0, x1w = (float)x1;
    const float y0w = (float)y0, y1w = (float)y1;
    const float wa = (x1w - x) * (y1w - y);
    const float wb = (x1w - x) * (y - y0w);
    const float wc = (x - x0w) * (y1w - y);
    const float wd = (x - x0w) * (y - y0w);

    const int    dx   = x1 - x0;                      // 0 (clamped) or 1
    const size_t idx0 = (size_t)((y0 << 8) | x0);     // y0*W + x0
    const size_t idx1 = (size_t)((y1 << 8) | x0);     // y1*W + x0

    const float* __restrict__ Ub = U   + (size_t)b * C_ * PLANE_;
    float*       __restrict__ Ob = out + (size_t)b * C_ * N_ + n;

#pragma unroll 8
    for (int c = 0; c < C_; ++c) {
        const float* p = Ub + (size_t)c * PLANE_;

        if (c + 1 < C_) {
            // Prefetch the next channel plane's two source rows.
            __builtin_prefetch(p + PLANE_ + idx0, 0, 3);
            __builtin_prefetch(p + PLANE_ + idx1, 0, 3);
        }

        const float Ia = p[idx0];
        const float Ic = p[idx0 + dx];
        const float Ib = p[idx1];
        const float Id = p[idx1 + dx];

        const float v = fmaf(wa, Ia, fmaf(wb, Ib, fmaf(wc, Ic, wd * Id)));

        // Streaming NT store: output is write-once; keep U resident in L2.
        __builtin_nontemporal_store(v, Ob + (size_t)c * N_);
    }
}

extern "C" void kernel_launch(void* const* d_in, const int* in_sizes, int n_in,
                              void* d_out, int out_size, void* d_ws, size_t ws_size,
                              hipStream_t stream) {
    (void)in_sizes; (void)n_in; (void)ws_size; (void)d_ws; (void)out_size;
    const float* U     = (const float*)d_in[0];
    const float* theta = (const float*)d_in[1];
    // d_in[2]/d_in[3] are out_h/out_w scalars (== 256); shapes are hardcoded.
    float* out = (float*)d_out;

    dim3 grid(HW_, B_);   // 256 rows x 8 batches = 2048 blocks
    dim3 block(256);      // 8 wave32s; one thread per output column
    st_bilinear_kernel<<<grid, block, 0, stream>>>(U, theta, out);
}